// SpikeFP64Round_43860206027302
// MI455X (gfx1250) — compile-verified
//
#include <hip/hip_runtime.h>
#include <stdint.h>

// ---------------------------------------------------------------------------
// SpikeFP64Round for MI455X (gfx1250, wave32)
//
// Memory-bound streaming op (1 GiB traffic, ~46us floor at 23.3 TB/s HBM).
//   * TDM (tensor_load_to_lds) double-buffered 4KB tiles per wave (TENSORcnt).
//   * Dynamic LDS (base offset 0) accessed via address_space(3) -> ds_load.
//   * ballot_w32 + v_bfrev_b32 pack 64 bit-floats -> uint64 in 4 instructions.
//   * v_add_f64 + v_floor_f64 (wave-uniform result).
//   * Result is wave-uniform => any lane extracts any bit: one nontemporal
//     global_store_b128 per ROW PAIR (512B/wave/instr, fully coalesced).
// ---------------------------------------------------------------------------

typedef __attribute__((ext_vector_type(4))) unsigned int v4u;
typedef __attribute__((ext_vector_type(8))) int          v8i;
typedef __attribute__((ext_vector_type(4))) int          v4i;
typedef __attribute__((ext_vector_type(4))) float        f32x4;
typedef __attribute__((address_space(3)))  float         lds_f32;

#define WAVES_PER_BLOCK 8
#define ROWS_PER_WAVE   16                                    // 16 rows * 256B = 4KB tile
#define ROWS_PER_BLOCK_TILE (WAVES_PER_BLOCK * ROWS_PER_WAVE) // 128 rows / block tile
#define TILE_DW    (ROWS_PER_WAVE * 64)                       // 1024 dwords per wave tile
#define TILE_BYTES (TILE_DW * 4)                              // 4096
#define WAVE_LDS_BYTES (2 * TILE_BYTES)                       // double buffered = 8KB
#define BLOCK_LDS_BYTES (WAVES_PER_BLOCK * WAVE_LDS_BYTES)    // 64KB dynamic LDS
#define MAX_BLOCKS 2048

// Issue one 1-D TDM copy: TILE_DW dwords from gsrc -> LDS byte offset lds_byte_off.
// Descriptor layout per CDNA5 ISA (D# groups).
__device__ __forceinline__ void tdm_load_tile(const float* gsrc, unsigned lds_byte_off) {
  unsigned long long ga = (unsigned long long)(size_t)gsrc;
  v4u g0;
  g0.x = 1u;                                             // count=1 (valid), user mode
  g0.y = lds_byte_off;                                   // lds_addr (bytes)
  g0.z = (unsigned)ga;                                   // global_addr[31:0]
  g0.w = (unsigned)((ga >> 32) & 0x01FFFFFFull)          // global_addr[56:32]
         | (2u << 30);                                   // type = 2 ("image")
  v8i g1;
  g1[0] = (int)(2u << 16);                               // wg_mask=0, data_size=4B
  g1[1] = (int)((TILE_DW & 0xFFFFu) << 16);              // tensor_dim0[15:0]
  g1[2] = (int)(((unsigned)TILE_DW >> 16) | (1u << 16)); // tensor_dim0[31:16] | tensor_dim1=1
  g1[3] = (int)(((unsigned)TILE_DW) << 16);              // tensor_dim1 hi=0 | tile_dim0
  g1[4] = 1;                                             // tile_dim1=1, tile_dim2=0
  g1[5] = (int)TILE_DW;                                  // tensor_dim0_stride[31:0]
  g1[6] = 0;
  g1[7] = 0;
  v4i gz = {0, 0, 0, 0};
#if defined(__clang_major__) && (__clang_major__ >= 23)
  v8i gz8 = {0, 0, 0, 0, 0, 0, 0, 0};
  __builtin_amdgcn_tensor_load_to_lds(g0, g1, gz, gz, gz8, 0);
#else
  __builtin_amdgcn_tensor_load_to_lds(g0, g1, gz, gz, 0);
#endif
}

// Pack one row's two 32-element halves (lane l holds elements l and 32+l) into
// the rounded result's hi/lo dwords (wave-uniform), sign bit restored.
__device__ __forceinline__ void pack_round(float hi_f, float lo_f,
                                           unsigned& o_hi, unsigned& o_lo) {
  unsigned u_hi = __builtin_bitreverse32(__builtin_amdgcn_ballot_w32(hi_f != 0.0f));
  unsigned u_lo = __builtin_bitreverse32(__builtin_amdgcn_ballot_w32(lo_f != 0.0f));
  unsigned sign = u_hi & 0x80000000u;
  unsigned long long uv =
      ((unsigned long long)(u_hi & 0x7FFFFFFFu) << 32) | (unsigned long long)u_lo;
  double res = __builtin_floor(__builtin_bit_cast(double, uv) + 0.5); // v_add_f64+v_floor_f64
  unsigned long long rb = __builtin_bit_cast(unsigned long long, res);
  o_hi = ((unsigned)(rb >> 32)) | sign;                  // res >= 0 => bit31 free
  o_lo = (unsigned)rb;
}

__global__ __launch_bounds__(256) void SpikeFP64Round_43860206027302_kernel(
    const float* __restrict__ x, float* __restrict__ out, int ntiles) {
  const unsigned lane   = threadIdx.x & 31u;
  const unsigned wave   = (unsigned)__builtin_amdgcn_readfirstlane((int)(threadIdx.x >> 5));
  const unsigned wlds   = wave * WAVE_LDS_BYTES;         // this wave's LDS byte base
  const unsigned rowsel = lane >> 4;                     // 0: even row, 1: odd row
  const unsigned q      = lane & 15u;                    // float4 slot within row
  const unsigned bb     = 31u - 4u * (lane & 7u);        // top bit index within dword

  const int t0 = (int)blockIdx.x;
  const int stride = (int)gridDim.x;

  if (t0 < ntiles) {
    const float* src0 =
        x + ((size_t)t0 * ROWS_PER_BLOCK_TILE + (size_t)wave * ROWS_PER_WAVE) * 64;
    tdm_load_tile(src0, wlds);                           // prologue -> buffer 0
  }

  unsigned buf = 0;
  for (int t = t0; t < ntiles; t += stride) {
    const int tn = t + stride;
    if (tn < ntiles) {
      const float* srcn =
          x + ((size_t)tn * ROWS_PER_BLOCK_TILE + (size_t)wave * ROWS_PER_WAVE) * 64;
      tdm_load_tile(srcn, wlds + (buf ^ 1u) * TILE_BYTES);
      __builtin_amdgcn_s_wait_tensorcnt(1);              // tile t landed; t+1 in flight
    } else {
      __builtin_amdgcn_s_wait_tensorcnt(0);
    }
    __asm__ __volatile__("" ::: "memory");

    const unsigned base = wlds + buf * TILE_BYTES;
    const size_t row0 = (size_t)t * ROWS_PER_BLOCK_TILE + (size_t)wave * ROWS_PER_WAVE;
#pragma unroll
    for (int k = 0; k < ROWS_PER_WAVE / 2; ++k) {
      // ds_load_b32 x4 (pairs mergeable into ds_load_2addr_b32), conflict-free
      const volatile lds_f32* p = (const volatile lds_f32*)(base + (unsigned)(2 * k) * 256u);
      float hiA = p[lane];
      float loA = p[32u + lane];
      float hiB = p[64u + lane];
      float loB = p[96u + lane];

      unsigned hA, lA, hB, lB;
      pack_round(hiA, loA, hA, lA);                      // row 2k   (wave-uniform)
      pack_round(hiB, loB, hB, lB);                      // row 2k+1 (wave-uniform)

      // lane picks its row and its dword, then extracts 4 consecutive bits.
      unsigned hsel = rowsel ? hB : hA;
      unsigned lsel = rowsel ? lB : lA;
      unsigned word = (lane & 8u) ? lsel : hsel;
      f32x4 o;
      o.x = (float)((word >> bb) & 1u);
      o.y = (float)((word >> (bb - 1u)) & 1u);
      o.z = (float)((word >> (bb - 2u)) & 1u);
      o.w = (float)((word >> (bb - 3u)) & 1u);

      float* drow = out + (row0 + (size_t)(2 * k) + (size_t)rowsel) * 64 + 4u * q;
      __builtin_nontemporal_store(o, (f32x4*)drow);      // 512B/wave b128 store, NT
    }
    __asm__ __volatile__("" ::: "memory");
    buf ^= 1u;
  }
}

// Fallback for rows not covered by full 128-row block tiles (direct loads).
__global__ __launch_bounds__(256) void SpikeFP64Round_tail_kernel(
    const float* __restrict__ x, float* __restrict__ out,
    long long row_start, long long rows) {
  const unsigned lane = threadIdx.x & 31u;
  long long row = row_start +
      (long long)blockIdx.x * (long long)(blockDim.x >> 5) + (long long)(threadIdx.x >> 5);
  if (row >= rows) return;                               // wave-uniform branch
  const float* irow = x + row * 64;
  unsigned r_hi, r_lo;
  pack_round(irow[lane], irow[32u + lane], r_hi, r_lo);
  unsigned shift = 31u - lane;
  float* orow = out + row * 64;
  orow[lane]       = (float)((r_hi >> shift) & 1u);
  orow[32u + lane] = (float)((r_lo >> shift) & 1u);
}

extern "C" void kernel_launch(void* const* d_in, const int* in_sizes, int n_in,
                              void* d_out, int out_size, void* d_ws, size_t ws_size,
                              hipStream_t stream) {
  const float* x = (const float*)d_in[0];
  float* out = (float*)d_out;
  const long long n = (long long)in_sizes[0];
  const long long rows = n / 64;
  const int ntiles = (int)(rows / ROWS_PER_BLOCK_TILE);
  if (ntiles > 0) {
    int grid = ntiles < MAX_BLOCKS ? ntiles : MAX_BLOCKS;
    SpikeFP64Round_43860206027302_kernel<<<grid, 256, BLOCK_LDS_BYTES, stream>>>(x, out, ntiles);
  }
  const long long covered = (long long)ntiles * ROWS_PER_BLOCK_TILE;
  const long long tail = rows - covered;
  if (tail > 0) {
    int tgrid = (int)((tail + 7) / 8);                   // 8 waves per 256-thread block
    SpikeFP64Round_tail_kernel<<<tgrid, 256, 0, stream>>>(x, out, covered, rows);
  }
}